// GSPExtendedKalmanNetNN_50749333569547
// MI455X (gfx1250) — compile-verified
//
#include <hip/hip_runtime.h>
#include <math.h>

// ---------------------------------------------------------------------------
// GSP Extended KalmanNet forward, MI455X (gfx1250).
// Roofline: ~1.2 GB of fp32 weights streamed once -> ~52us at 23.3 TB/s HBM.
// Pure bandwidth-bound GEMV; MACs executed on V_WMMA_F32_16X16X4_F32.
// Weight loads are non-temporal (single-use stream >> 192MB L2).
// ---------------------------------------------------------------------------

#define MDIM 256
#define DIN  768
#define H1D  4096
#define HID  5120
#define G3   (3 * HID)   // 15360
#define H2D  1024
#define EPSN 1e-12f

typedef __attribute__((ext_vector_type(2))) float v2f;
typedef __attribute__((ext_vector_type(8))) float v8f;

// ---------------------------------------------------------------------------
// Wave-level WMMA GEMV: computes dot(W[r][0..K), x) for 16 consecutive rows.
// W: pointer to row 0 of the 16-row tile (row stride = ldw floats, may be a
//    K-chunk view). x: LDS pointer, K floats, K % 8 == 0.
// Returns: every lane holds the result for row (lane & 15).
//
// Layouts (CDNA5 ISA 7.12.2, f32 16x16x4):
//   A (16x4): VGPR0 = K0 (lanes 0-15) / K2 (lanes 16-31); VGPR1 = K1 / K3.
//     We broadcast x[k0..k0+3] into all 16 A rows -> all D rows identical.
//   B (4x16): VGPR0 = rows K0 / K2 striped over lanes (N = lane % 16);
//     VGPR1 = rows K1 / K3. Weights go into B so outputs sit on the N axis.
//   D (16x16): VGPR0, lanes 0-15 = (M=0, N=lane); acc[0] is the result.
//
// Each lane loads exactly the two weight floats it owns per WMMA with a
// non-temporal global_load_b64 at row + k + 2*hi (lanes l / l+16 together
// cover 16 consecutive bytes of each row) -> no cross-lane shuffles, no
// LDS-pipe traffic for weights, 2 loads + 2 WMMAs per 8 K-columns.
// ---------------------------------------------------------------------------
__device__ __forceinline__ float wmma_gemv16(const float* __restrict__ W, int ldw,
                                             const float* __restrict__ xs, int K,
                                             int lane)
{
    v8f acc = {0.f, 0.f, 0.f, 0.f, 0.f, 0.f, 0.f, 0.f};
    const int hi = lane >> 4;  // 0: lanes 0-15, 1: lanes 16-31
    const float* wrow = W + (size_t)(lane & 15) * (size_t)ldw + 2 * hi;
    const float* xh = xs + 2 * hi;
#pragma unroll 8
    for (int k0 = 0; k0 < K; k0 += 8) {
        v2f b0 = __builtin_nontemporal_load((const v2f*)(wrow + k0));      // K=k0..k0+3
        v2f b1 = __builtin_nontemporal_load((const v2f*)(wrow + k0 + 4));  // K=k0+4..k0+7
        v2f a0, a1;
        a0[0] = xh[k0 + 0]; a0[1] = xh[k0 + 1];     // A quad k0..k0+3 (bcast)
        a1[0] = xh[k0 + 4]; a1[1] = xh[k0 + 5];     // A quad k0+4..k0+7
        acc = __builtin_amdgcn_wmma_f32_16x16x4_f32(false, a0, false, b0,
                                                    (short)0, acc, false, false);
        acc = __builtin_amdgcn_wmma_f32_16x16x4_f32(false, a1, false, b1,
                                                    (short)0, acc, false, false);
    }
    return acc[0];
}

// ---------------------------------------------------------------------------
// Chunked GEMV: grid.x * 8 waves cover row tiles of 16, grid.y covers K
// chunks. Partials stored to part[chunk][nrows] (no atomics -> deterministic).
// ---------------------------------------------------------------------------
__global__ __launch_bounds__(256) void gemv16_part(const float* __restrict__ W, int K,
                                                   const float* __restrict__ x, int kchunk,
                                                   float* __restrict__ part, int nrows)
{
    __shared__ float xs[1024];
    const int kc = blockIdx.y * kchunk;
    for (int i = threadIdx.x; i < kchunk; i += blockDim.x) xs[i] = x[kc + i];
    __syncthreads();
    const int lane = threadIdx.x & 31;
    const int wave = threadIdx.x >> 5;
    const int r0 = (blockIdx.x * 8 + wave) * 16;
    const float* Wt = W + (size_t)r0 * (size_t)K + kc;
    float v = wmma_gemv16(Wt, K, xs, kchunk, lane);
    if (lane < 16) part[(size_t)blockIdx.y * (size_t)nrows + r0 + lane] = v;
}

__global__ void bias_relu_sum(const float* __restrict__ part, int nchunk, int n,
                              const float* __restrict__ b, float* __restrict__ out)
{
    int i = blockIdx.x * blockDim.x + threadIdx.x;
    if (i >= n) return;
    float s = b[i];
    for (int c = 0; c < nchunk; ++c) s += part[(size_t)c * n + i];
    out[i] = fmaxf(s, 0.f);
}

// PyTorch GRU cell: r=sig(ir+hr), z=sig(iz+hz), g=tanh(ig + r*hg),
// h' = (1-z)*g + z*h. gi/gh partials summed here (biases included).
__global__ void gru_combine(const float* __restrict__ giP, int nci,
                            const float* __restrict__ ghP, int nch,
                            const float* __restrict__ bih, const float* __restrict__ bhh,
                            const float* __restrict__ hprev, float* __restrict__ hout)
{
    int j = blockIdx.x * blockDim.x + threadIdx.x;
    if (j >= HID) return;
    float ir = bih[j], iz = bih[HID + j], ig = bih[2 * HID + j];
    for (int c = 0; c < nci; ++c) {
        const float* p = giP + (size_t)c * G3;
        ir += p[j]; iz += p[HID + j]; ig += p[2 * HID + j];
    }
    float hr = bhh[j], hz = bhh[HID + j], hg = bhh[2 * HID + j];
    for (int c = 0; c < nch; ++c) {
        const float* p = ghP + (size_t)c * G3;
        hr += p[j]; hz += p[HID + j]; hg += p[2 * HID + j];
    }
    float r = 1.f / (1.f + __expf(-(ir + hr)));
    float z = 1.f / (1.f + __expf(-(iz + hz)));
    float g = tanhf(ig + r * hg);
    hout[j] = (1.f - z) * g + z * hprev[j];
}

// ---------------------------------------------------------------------------
// Front end: all 256-wide chained matvecs + normalize/concat, one workgroup
// (512 thr = 16 waves = 16 row tiles per matvec), WMMA GEMV throughout.
// ---------------------------------------------------------------------------
__global__ void __launch_bounds__(512) front_kernel(
    const float* __restrict__ y, const float* __restrict__ V,
    const float* __restrict__ Vt, const float* __restrict__ F,
    const float* __restrict__ Hm,
    const float* __restrict__ m1x_post, const float* __restrict__ m1x_prev,
    const float* __restrict__ state0,
    float* __restrict__ kin, float* __restrict__ yt_out,
    float* __restrict__ m1y_out, float* __restrict__ xprior_out)
{
    __shared__ float sA[256], sB[256], sMP[256], sYT[256], sOBS[256], red[256];
    const int t = threadIdx.x;
    const int lane = t & 31;
    const int r0 = (t >> 5) * 16;   // wave -> row tile

    auto mv = [&](const float* A, const float* src, float* dst) {
        float v = wmma_gemv16(A + (size_t)r0 * 256, 256, src, 256, lane);
        if (lane < 16) dst[r0 + lane] = v;
        __syncthreads();
    };
    auto sumsq = [&](const float* v) -> float {
        if (t < 256) red[t] = v[t] * v[t];
        __syncthreads();
        for (int s = 128; s > 0; s >>= 1) {
            if (t < s) red[t] += red[t + s];
            __syncthreads();
        }
        float r = red[0];
        __syncthreads();
        return r;
    };

    if (t < 256) sMP[t] = m1x_post[t];
    __syncthreads();

    // x_prior = V_t @ (F @ (V @ m1x_post))
    mv(V,  sMP, sA);
    mv(F,  sA,  sB);
    mv(Vt, sB,  sA);
    if (t < 256) xprior_out[t] = sA[t];
    // m1y = V_t @ (H @ (V @ x_prior))
    mv(V,  sA,  sB);
    mv(Hm, sB,  sA);
    mv(Vt, sA,  sB);
    if (t < 256) m1y_out[t] = sB[t];
    // yt = V_t @ y
    if (t < 256) sA[t] = y[t];
    __syncthreads();
    mv(Vt, sA, sYT);
    if (t < 256) yt_out[t] = sYT[t];
    // obs0 = H @ (F @ state0)
    if (t < 256) sA[t] = state0[t];
    __syncthreads();
    mv(F,  sA, sB);
    mv(Hm, sB, sOBS);

    // kin = [normalize(yt-obs0), normalize(m1x-m1x_prev), normalize(m1x)]
    if (t < 256) sA[t] = sYT[t] - sOBS[t];
    __syncthreads();
    float n1 = sumsq(sA);
    if (t < 256) kin[t] = sA[t] * (1.f / fmaxf(sqrtf(n1), EPSN));

    if (t < 256) sB[t] = sMP[t] - m1x_prev[t];
    __syncthreads();
    float n3 = sumsq(sB);
    if (t < 256) kin[256 + t] = sB[t] * (1.f / fmaxf(sqrtf(n3), EPSN));

    float n4 = sumsq(sMP);
    if (t < 256) kin[512 + t] = sMP[t] * (1.f / fmaxf(sqrtf(n4), EPSN));
}

// ---------------------------------------------------------------------------
// Back end: l2 = relu(sum l2P + b2); kg = W3@l2 + b3; x_post = x_prior +
// kg*(yt - m1y); out = V @ x_post. One workgroup (256 thr = 8 waves).
// ---------------------------------------------------------------------------
__global__ void __launch_bounds__(256) final_kernel(
    const float* __restrict__ l2P, int nchunk,
    const float* __restrict__ b2, const float* __restrict__ W3,
    const float* __restrict__ b3, const float* __restrict__ V,
    const float* __restrict__ xprior, const float* __restrict__ yt,
    const float* __restrict__ m1y, float* __restrict__ out)
{
    __shared__ float l2s[H2D];
    __shared__ float xp[256];
    const int t = threadIdx.x;
    const int lane = t & 31;
    const int wave = t >> 5;

    for (int i = t; i < H2D; i += 256) {
        float s = b2[i];
        for (int c = 0; c < nchunk; ++c) s += l2P[(size_t)c * H2D + i];
        l2s[i] = fmaxf(s, 0.f);
    }
    __syncthreads();

    for (int tt = wave; tt < 16; tt += 8) {           // uniform per wave
        int r0 = tt * 16;
        float kg = wmma_gemv16(W3 + (size_t)r0 * H2D, H2D, l2s, H2D, lane);
        if (lane < 16) {
            int r = r0 + lane;
            float kgv = kg + b3[r];
            xp[r] = xprior[r] + kgv * (yt[r] - m1y[r]);
        }
    }
    __syncthreads();

    for (int tt = wave; tt < 16; tt += 8) {
        int r0 = tt * 16;
        float v = wmma_gemv16(V + (size_t)r0 * 256, 256, xp, 256, lane);
        if (lane < 16) out[r0 + lane] = v;
    }
}

// ---------------------------------------------------------------------------
extern "C" void kernel_launch(void* const* d_in, const int* in_sizes, int n_in,
                              void* d_out, int out_size, void* d_ws, size_t ws_size,
                              hipStream_t stream)
{
    const float* y    = (const float*)d_in[0];
    const float* V    = (const float*)d_in[1];
    const float* Vt   = (const float*)d_in[2];
    const float* F    = (const float*)d_in[3];
    const float* Hm   = (const float*)d_in[4];
    const float* m1x  = (const float*)d_in[5];
    const float* m1xp = (const float*)d_in[6];
    const float* st0  = (const float*)d_in[7];
    const float* hn   = (const float*)d_in[8];   // (2,1,HID)
    const float* W1   = (const float*)d_in[9];
    const float* b1   = (const float*)d_in[10];
    const float* Wih0 = (const float*)d_in[11];
    const float* Whh0 = (const float*)d_in[12];
    const float* bih0 = (const float*)d_in[13];
    const float* bhh0 = (const float*)d_in[14];
    const float* Wih1 = (const float*)d_in[15];
    const float* Whh1 = (const float*)d_in[16];
    const float* bih1 = (const float*)d_in[17];
    const float* bhh1 = (const float*)d_in[18];
    const float* W2   = (const float*)d_in[19];
    const float* b2   = (const float*)d_in[20];
    const float* W3   = (const float*)d_in[21];
    const float* b3   = (const float*)d_in[22];
    float* out = (float*)d_out;
    float* ws  = (float*)d_ws;

    // Workspace layout (floats). Partials are per-K-chunk (deterministic sum).
    float* l1P  = ws;                  // 3  * H1D
    float* gi0P = l1P  + 3  * H1D;     // 4  * G3
    float* gh0P = gi0P + 4  * G3;      // 5  * G3
    float* gi1P = gh0P + 5  * G3;      // 5  * G3
    float* gh1P = gi1P + 5  * G3;      // 5  * G3
    float* l2P  = gh1P + 5  * G3;      // 10 * H2D
    float* kin  = l2P  + 10 * H2D;     // DIN
    float* yt   = kin  + DIN;          // 256
    float* m1y  = yt   + MDIM;         // 256
    float* xpr  = m1y  + MDIM;         // 256
    float* l1   = xpr  + MDIM;         // H1D
    float* h0   = l1   + H1D;          // HID
    float* h1   = h0   + HID;          // HID  (total ~1.26 MB)

    const float* hn0 = hn;
    const float* hn1 = hn + HID;

    // Stage 1: spectral matvec chains + feature build.
    front_kernel<<<1, 512, 0, stream>>>(y, V, Vt, F, Hm, m1x, m1xp, st0,
                                        kin, yt, m1y, xpr);

    // l1 = relu(W1 @ kin + b1): 4096 rows, K=768 in 3 chunks of 256.
    gemv16_part<<<dim3(H1D / 128, 3), 256, 0, stream>>>(W1, DIN, kin, 256, l1P, H1D);
    bias_relu_sum<<<H1D / 256, 256, 0, stream>>>(l1P, 3, H1D, b1, l1);

    // GRU layer 0: gi = Wih0@l1, gh = Whh0@hn0 (3840/4800 waves in flight).
    gemv16_part<<<dim3(G3 / 128, 4), 256, 0, stream>>>(Wih0, H1D, l1,  1024, gi0P, G3);
    gemv16_part<<<dim3(G3 / 128, 5), 256, 0, stream>>>(Whh0, HID, hn0, 1024, gh0P, G3);
    gru_combine<<<HID / 256, 256, 0, stream>>>(gi0P, 4, gh0P, 5, bih0, bhh0, hn0, h0);

    // GRU layer 1.
    gemv16_part<<<dim3(G3 / 128, 5), 256, 0, stream>>>(Wih1, HID, h0,  1024, gi1P, G3);
    gemv16_part<<<dim3(G3 / 128, 5), 256, 0, stream>>>(Whh1, HID, hn1, 1024, gh1P, G3);
    gru_combine<<<HID / 256, 256, 0, stream>>>(gi1P, 5, gh1P, 5, bih1, bhh1, hn1, h1);

    // l2 partials: 1024 rows, K=5120 in 10 chunks of 512.
    gemv16_part<<<dim3(H2D / 128, 10), 256, 0, stream>>>(W2, HID, h1, 512, l2P, H2D);

    // kg, posterior update, IGFT.
    final_kernel<<<1, 256, 0, stream>>>(l2P, 10, b2, W3, b3, V, xpr, yt, m1y, out);
}